// MatchingLoss_73504070304261
// MI455X (gfx1250) — compile-verified
//
#include <hip/hip_runtime.h>
#include <stdint.h>
#include <math.h>

#define NMAT 27
#define NN   (NMAT * NMAT)
#define WAVES_PER_BLOCK 8
#define BLOCK_THREADS   (WAVES_PER_BLOCK * 32)

#if __has_builtin(__builtin_amdgcn_global_load_async_to_lds_b32) && \
    __has_builtin(__builtin_amdgcn_s_wait_asynccnt)
#define USE_ASYNC_LDS 1
typedef __attribute__((address_space(1))) int as1_int;
typedef __attribute__((address_space(3))) int as3_int;
#endif

// One wave32 per batch sample. Lane L represents JV column index j=L (j=0..27
// are the 1-based algorithm's columns incl. the virtual column 0); lanes 28-31 idle.
__global__ __launch_bounds__(BLOCK_THREADS)
void hungarian_ce_kernel(const float* __restrict__ predicted,
                         const int*   __restrict__ target,
                         double*      __restrict__ partials,
                         int B)
{
    __shared__ float  s_cost[WAVES_PER_BLOCK][NN];       // probs for this wave's sample
    __shared__ float  s_u[WAVES_PER_BLOCK][NMAT + 1];    // row potentials u[0..27]
    __shared__ double s_part[WAVES_PER_BLOCK];

    const int lane = threadIdx.x & 31;
    const int wid  = threadIdx.x >> 5;
    const int b    = blockIdx.x * WAVES_PER_BLOCK + wid;
    const bool active = (b < B);

    // ---- stage the 27x27 probs matrix into LDS (async-to-LDS on CDNA5) ----
    if (active) {
        const float* g = predicted + (size_t)b * NN;
#ifdef USE_ASYNC_LDS
        for (int k = lane; k < NN; k += 32) {
            __builtin_amdgcn_global_load_async_to_lds_b32(
                (as1_int*)(uintptr_t)(g + k),
                (as3_int*)(uintptr_t)(&s_cost[wid][k]),
                0, 0);
        }
        __builtin_amdgcn_s_wait_asynccnt(0);
#else
        for (int k = lane; k < NN; k += 32) s_cost[wid][k] = g[k];
#endif
    }
    if (lane <= NMAT) s_u[wid][lane] = 0.0f;
    __syncthreads();

    float v    = 0.0f;   // column potential v[lane]
    float minv = 0.0f;
    int   p    = 0;      // p[lane]: row (1-based) matched to column lane
    int   way  = 0;
    float contrib = 0.0f;

    if (active) {
        const bool isCol = (lane >= 1 && lane <= NMAT);
        for (int i = 1; i <= NMAT; ++i) {
            minv = INFINITY;
            bool used = false;
            if (lane == 0) p = i;        // p[0] = i
            int j0 = 0;
            for (;;) {
                if (lane == j0) used = true;
                int   i0   = __shfl(p, j0);        // row on the path (1..27)
                float u_i0 = s_u[wid][i0];
                if (isCol && !used) {
                    // cost = -probs (min-cost == max-weight matching)
                    float cur = -s_cost[wid][(i0 - 1) * NMAT + (lane - 1)] - u_i0 - v;
                    if (cur < minv) { minv = cur; way = j0; }
                }
                // masked argmin over free real columns, np.argmin tie-break (lowest j)
                float best = (isCol && !used) ? minv : INFINITY;
                int   bj   = lane;
                #pragma unroll
                for (int off = 16; off > 0; off >>= 1) {
                    float ov = __shfl_xor(best, off);
                    int   oj = __shfl_xor(bj,   off);
                    if (ov < best || (ov == best && oj < bj)) { best = ov; bj = oj; }
                }
                float delta = best;
                int   j1    = bj;
                if (used && lane <= NMAT) {
                    s_u[wid][p] += delta;          // u[p[j]] += delta (distinct rows)
                    v -= delta;                     // v[j]   -= delta
                } else if (isCol) {
                    minv -= delta;                  // minv[free] -= delta
                }
                __builtin_amdgcn_wave_barrier();    // keep LDS RMW ordered vs next read
                j0 = j1;
                if (__shfl(p, j0) == 0) break;      // reached a free column
            }
            // augment: while j0: j1=way[j0]; p[j0]=p[j1]; j0=j1
            int j = j0;
            while (j != 0) {
                int jprev = __shfl(way, j);
                int pv    = __shfl(p, jprev);
                if (lane == j) p = pv;
                j = jprev;
            }
        }

        // ---- masked cross-entropy: lane j handles column c=j-1 ----
        if (isCol) {
            int   c    = lane - 1;
            int   r    = p - 1;                              // matched row for col c
            float mval = s_cost[wid][r * NMAT + c];
            int   t    = target[(size_t)b * NMAT + c];
            // column of res = {0 x26, mval}: lse = log(26 + exp(mval)), stably
            float mx  = fmaxf(mval, 0.0f);
            float lse = mx + logf(26.0f * expf(-mx) + expf(mval - mx));
            float res_t = (t == r) ? mval : 0.0f;
            contrib = lse - res_t;
        }
    }

    // ---- deterministic two-level reduction ----
    float wsum = contrib;
    #pragma unroll
    for (int off = 16; off > 0; off >>= 1) wsum += __shfl_xor(wsum, off);
    if (lane == 0) s_part[wid] = (double)wsum;
    __syncthreads();
    if (threadIdx.x == 0) {
        double tot = 0.0;
        #pragma unroll
        for (int w = 0; w < WAVES_PER_BLOCK; ++w) tot += s_part[w];
        partials[blockIdx.x] = tot;
    }
}

__global__ __launch_bounds__(256)
void reduce_partials_kernel(const double* __restrict__ partials, int nparts,
                            float* __restrict__ out, double inv_count)
{
    __shared__ double s[256];
    double acc = 0.0;
    for (int i = threadIdx.x; i < nparts; i += 256) acc += partials[i];
    s[threadIdx.x] = acc;
    __syncthreads();
    for (int off = 128; off > 0; off >>= 1) {
        if ((int)threadIdx.x < off) s[threadIdx.x] += s[threadIdx.x + off];
        __syncthreads();
    }
    if (threadIdx.x == 0) out[0] = (float)(s[0] * inv_count);
}

extern "C" void kernel_launch(void* const* d_in, const int* in_sizes, int n_in,
                              void* d_out, int out_size, void* d_ws, size_t ws_size,
                              hipStream_t stream)
{
    const float* predicted = (const float*)d_in[0];   // [B, 729] f32
    const int*   target    = (const int*)d_in[1];     // [B, 27]  i32
    const int B = in_sizes[0] / NN;
    const int nblocks = (B + WAVES_PER_BLOCK - 1) / WAVES_PER_BLOCK;

    double* partials = (double*)d_ws;                 // nblocks doubles

    hungarian_ce_kernel<<<nblocks, BLOCK_THREADS, 0, stream>>>(
        predicted, target, partials, B);

    const double inv_count = 1.0 / ((double)B * (double)NMAT);
    reduce_partials_kernel<<<1, 256, 0, stream>>>(
        partials, nblocks, (float*)d_out, inv_count);
}